// GCN2_80676665688552
// MI455X (gfx1250) — compile-verified
//
#include <hip/hip_runtime.h>
#include <hip/hip_bf16.h>

// ---------------------------------------------------------------------------
// GCN 2-layer forward for MI455X (gfx1250, wave32).
// GEMMs use V_WMMA_F32_16X16X4_F32 (fp32 in/out).
// Edge scatter uses coalesced float4 gathers + global_atomic_add_f32.
// ---------------------------------------------------------------------------

#define N_NODES 100000
#define NUM_E   1600000
#define IN_F    128
#define HID_F   64
#define OUT_F   40
#define OUT_P   48          // OUT padded to 3x16 WMMA tiles
#define BN_EPS  1e-5f

typedef __attribute__((ext_vector_type(2))) float v2f;
typedef __attribute__((ext_vector_type(8))) float v8f;

// ---------------- utility -------------------------------------------------

__global__ void k_zero_f32(float* __restrict__ p, long n) {
    long i = (long)blockIdx.x * blockDim.x + threadIdx.x;
    if (i < n) p[i] = 0.0f;
}

// degree count at dst (float so we can reuse buffer for rsqrt in place)
__global__ void k_count_deg(const long long* __restrict__ dst,
                            float* __restrict__ deg) {
    long e = (long)blockIdx.x * blockDim.x + threadIdx.x;
    if (e < NUM_E) atomicAdd(&deg[(int)dst[e]], 1.0f);
}

__global__ void k_dinv(float* __restrict__ deg) {
    long i = (long)blockIdx.x * blockDim.x + threadIdx.x;
    if (i < N_NODES) deg[i] = rsqrtf(deg[i] + 1.0f);
}

// ---------------- GEMM1: h = x @ W1   [N,128]x[128,64] --------------------
// Block: 256 threads = 8 waves. Waves tile 2(M) x 4(N) 16x16 tiles.
// W1 staged in LDS (32 KB). K-loop: 32 x wmma_f32_16x16x4_f32.
__global__ void __launch_bounds__(256)
k_gemm1(const float* __restrict__ x, const float* __restrict__ W1,
        float* __restrict__ h) {
    __shared__ float sW[IN_F * HID_F];           // 8192 floats = 32 KB
    int tid = threadIdx.x;
    for (int i = tid * 4; i < IN_F * HID_F; i += 256 * 4)
        *reinterpret_cast<float4*>(&sW[i]) =
            *reinterpret_cast<const float4*>(&W1[i]);
    __syncthreads();

    int wave = tid >> 5;
    int lane = tid & 31;
    int mt   = blockIdx.x * 2 + (wave >> 2);     // M tile (16 rows)
    int nt   = wave & 3;                          // N tile (16 cols)
    int row  = mt * 16 + (lane & 15);
    int koff = (lane >> 4) * 2;                   // 0 or 2
    int col  = nt * 16 + (lane & 15);

    const float* xrow = x + (long)row * IN_F;
    v8f c = {};
    #pragma unroll 4
    for (int k = 0; k < IN_F; k += 4) {
        v2f a = *reinterpret_cast<const v2f*>(&xrow[k + koff]);
        v2f b;
        b.x = sW[(k + koff)     * HID_F + col];
        b.y = sW[(k + koff + 1) * HID_F + col];
        c = __builtin_amdgcn_wmma_f32_16x16x4_f32(
                false, a, false, b, (short)0, c, false, false);
    }

    // C/D layout: VGPR r holds row (r + 8*(lane/16)), col = lane%16
    float* hp = h + (long)(mt * 16) * HID_F + nt * 16 + (lane & 15);
    int rbase = (lane >> 4) * 8;
    #pragma unroll
    for (int r = 0; r < 8; ++r)
        hp[(long)(rbase + r) * HID_F] = c[r];
}

// ---------------- self-loop + bias init: out = h*dinv^2 + bias ------------
__global__ void k_self_init(const float* __restrict__ hlin,
                            const float* __restrict__ dinv,
                            const float* __restrict__ bias,
                            float* __restrict__ out, int F) {
    long idx = (long)blockIdx.x * blockDim.x + threadIdx.x;
    if (idx >= (long)N_NODES * F) return;
    int i = (int)(idx / F), f = (int)(idx % F);
    float di = dinv[i];
    out[idx] = hlin[idx] * di * di + bias[f];
}

// ---------------- edge scatter, F=64: 16 lanes/edge, float4 each ----------
__global__ void k_edge_scatter64(const float* __restrict__ hlin,
                                 const long long* __restrict__ src,
                                 const long long* __restrict__ dst,
                                 const float* __restrict__ dinv,
                                 float* __restrict__ out) {
    long t = (long)blockIdx.x * blockDim.x + threadIdx.x;
    long e = t >> 4;
    if (e >= NUM_E) return;
    int f4 = (int)(t & 15) * 4;
    int s = (int)src[e], d = (int)dst[e];
    __builtin_prefetch(&src[e + 4096], 0, 1);
    float w = dinv[s] * dinv[d];
    float4 hv = *reinterpret_cast<const float4*>(&hlin[(long)s * HID_F + f4]);
    float* op = &out[(long)d * HID_F + f4];
    atomicAdd(op + 0, hv.x * w);
    atomicAdd(op + 1, hv.y * w);
    atomicAdd(op + 2, hv.z * w);
    atomicAdd(op + 3, hv.w * w);
}

// ---------------- BN: partial sums (sum, sumsq) per column ----------------
__global__ void __launch_bounds__(256)
k_bn_stats(const float* __restrict__ h, float* __restrict__ sums) {
    __shared__ float ssum[HID_F], ssq[HID_F];
    int tid = threadIdx.x;
    int f = tid & 63;
    int rphase = tid >> 6;                        // 0..3
    float s = 0.f, q = 0.f;
    for (long i = (long)blockIdx.x * 4 + rphase; i < N_NODES;
         i += (long)gridDim.x * 4) {
        float v = h[i * HID_F + f];
        s += v; q += v * v;
    }
    if (tid < HID_F) { ssum[tid] = 0.f; ssq[tid] = 0.f; }
    __syncthreads();
    atomicAdd(&ssum[f], s);                       // ds_add_f32
    atomicAdd(&ssq[f], q);
    __syncthreads();
    if (tid < HID_F) {
        atomicAdd(&sums[tid],          ssum[tid]);
        atomicAdd(&sums[HID_F + tid],  ssq[tid]);
    }
}

__global__ void k_bn_finalize(const float* __restrict__ sums,
                              const float* __restrict__ gamma,
                              const float* __restrict__ beta,
                              float* __restrict__ ss) {
    int f = threadIdx.x;
    if (f < HID_F) {
        float inv_n = 1.0f / (float)N_NODES;
        float mean  = sums[f] * inv_n;
        float var   = sums[HID_F + f] * inv_n - mean * mean;
        float sc    = gamma[f] * rsqrtf(var + BN_EPS);
        ss[f]          = sc;
        ss[HID_F + f]  = beta[f] - mean * sc;
    }
}

__global__ void k_bn_apply_relu(const float* __restrict__ in,
                                const float* __restrict__ ss,
                                float* __restrict__ outp) {
    long idx = (long)blockIdx.x * blockDim.x + threadIdx.x;
    if (idx >= (long)N_NODES * HID_F) return;
    int f = (int)(idx & 63);
    float v = in[idx] * ss[f] + ss[HID_F + f];
    outp[idx] = v > 0.f ? v : 0.f;
}

// ---------------- pad W2 [64,40] -> [64,48] -------------------------------
__global__ void k_pad_w2(const float* __restrict__ W2, float* __restrict__ W2p) {
    int idx = blockIdx.x * blockDim.x + threadIdx.x;
    if (idx >= HID_F * OUT_P) return;
    int k = idx / OUT_P, c = idx % OUT_P;
    W2p[idx] = (c < OUT_F) ? W2[k * OUT_F + c] : 0.0f;
}

// ---------------- GEMM2: y = hrelu @ W2p   [N,64]x[64,48] -----------------
// Block: 256 threads = 8 waves; each wave owns one 16-row strip and loops
// over 3 N-tiles. K-loop: 16 x wmma. Stores only cols < 40 (stride OUT_F).
__global__ void __launch_bounds__(256)
k_gemm2(const float* __restrict__ h, const float* __restrict__ W2p,
        float* __restrict__ y) {
    __shared__ float sW[HID_F * OUT_P];          // 3072 floats = 12 KB
    int tid = threadIdx.x;
    for (int i = tid * 4; i < HID_F * OUT_P; i += 256 * 4)
        *reinterpret_cast<float4*>(&sW[i]) =
            *reinterpret_cast<const float4*>(&W2p[i]);
    __syncthreads();

    int wave = tid >> 5;
    int lane = tid & 31;
    int mt = blockIdx.x * 8 + wave;               // 16-row tile index
    if (mt * 16 >= N_NODES) return;               // wave-uniform exit
    int row  = mt * 16 + (lane & 15);
    int koff = (lane >> 4) * 2;
    const float* hrow = h + (long)row * HID_F;

    for (int nt = 0; nt < 3; ++nt) {
        int col = nt * 16 + (lane & 15);
        v8f c = {};
        #pragma unroll 4
        for (int k = 0; k < HID_F; k += 4) {
            v2f a = *reinterpret_cast<const v2f*>(&hrow[k + koff]);
            v2f b;
            b.x = sW[(k + koff)     * OUT_P + col];
            b.y = sW[(k + koff + 1) * OUT_P + col];
            c = __builtin_amdgcn_wmma_f32_16x16x4_f32(
                    false, a, false, b, (short)0, c, false, false);
        }
        int rbase = (lane >> 4) * 8;
        if (col < OUT_F) {                        // mask padded columns
            #pragma unroll
            for (int r = 0; r < 8; ++r)
                y[(long)(mt * 16 + rbase + r) * OUT_F + col] = c[r];
        }
    }
}

// ---------------- edge scatter, F=40: 10 lanes/edge, float4 each ----------
__global__ void k_edge_scatter40(const float* __restrict__ hlin,
                                 const long long* __restrict__ src,
                                 const long long* __restrict__ dst,
                                 const float* __restrict__ dinv,
                                 float* __restrict__ out) {
    long t = (long)blockIdx.x * blockDim.x + threadIdx.x;
    long e = t / 10;
    if (e >= NUM_E) return;
    int f4 = (int)(t % 10) * 4;
    int s = (int)src[e], d = (int)dst[e];
    __builtin_prefetch(&dst[e + 4096], 0, 1);
    float w = dinv[s] * dinv[d];
    float4 hv = *reinterpret_cast<const float4*>(&hlin[(long)s * OUT_F + f4]);
    float* op = &out[(long)d * OUT_F + f4];
    atomicAdd(op + 0, hv.x * w);
    atomicAdd(op + 1, hv.y * w);
    atomicAdd(op + 2, hv.z * w);
    atomicAdd(op + 3, hv.w * w);
}

// ---------------------------------------------------------------------------

static inline size_t align256(size_t x) { return (x + 255) & ~(size_t)255; }

extern "C" void kernel_launch(void* const* d_in, const int* in_sizes, int n_in,
                              void* d_out, int out_size, void* d_ws, size_t ws_size,
                              hipStream_t stream) {
    (void)in_sizes; (void)n_in; (void)out_size; (void)ws_size;

    const float*     x     = (const float*)d_in[0];
    const long long* eidx  = (const long long*)d_in[1];   // int64 [2,E]
    const float*     W1    = (const float*)d_in[2];
    const float*     b1    = (const float*)d_in[3];
    const float*     gamma = (const float*)d_in[4];
    const float*     beta  = (const float*)d_in[5];
    const float*     W2    = (const float*)d_in[6];
    const float*     b2    = (const float*)d_in[7];
    float*           out   = (float*)d_out;

    const long long* src = eidx;
    const long long* dst = eidx + NUM_E;

    // workspace carve-up (dinv and sums each zeroed by their own pass)
    char* ws = (char*)d_ws;
    size_t off = 0;
    float* dinv = (float*)(ws + off); off += align256((size_t)N_NODES * 4);
    float* sums = (float*)(ws + off); off += align256(2 * HID_F * 4);
    float* ss   = (float*)(ws + off); off += align256(2 * HID_F * 4);
    float* W2p  = (float*)(ws + off); off += align256((size_t)HID_F * OUT_P * 4);
    float* bufA = (float*)(ws + off); off += align256((size_t)N_NODES * HID_F * 4);
    float* bufB = (float*)(ws + off); off += align256((size_t)N_NODES * HID_F * 4);

    const int B = 256;

    // 1) zero deg + BN sums (contiguous-ish: zero each region)
    k_zero_f32<<<(N_NODES + B - 1) / B, B, 0, stream>>>(dinv, N_NODES);
    k_zero_f32<<<1, B, 0, stream>>>(sums, 2 * HID_F);

    // 2) degree at dst, then dinv = rsqrt(deg+1)
    k_count_deg<<<(NUM_E + B - 1) / B, B, 0, stream>>>(dst, dinv);
    k_dinv<<<(N_NODES + B - 1) / B, B, 0, stream>>>(dinv);

    // 3) GEMM1: bufA = x @ W1     (N/32 = 3125 blocks, exact)
    k_gemm1<<<N_NODES / 32, 256, 0, stream>>>(x, W1, bufA);

    // 4) layer-1 aggregation into bufB
    long n64 = (long)N_NODES * HID_F;
    k_self_init<<<(unsigned)((n64 + B - 1) / B), B, 0, stream>>>(bufA, dinv, b1, bufB, HID_F);
    long t64 = (long)NUM_E * 16;
    k_edge_scatter64<<<(unsigned)((t64 + B - 1) / B), B, 0, stream>>>(bufA, src, dst, dinv, bufB);

    // 5) BatchNorm + ReLU: bufA = relu(bn(bufB))
    k_bn_stats<<<240, 256, 0, stream>>>(bufB, sums);
    k_bn_finalize<<<1, 64, 0, stream>>>(sums, gamma, beta, ss);
    k_bn_apply_relu<<<(unsigned)((n64 + B - 1) / B), B, 0, stream>>>(bufB, ss, bufA);

    // 6) GEMM2: bufB[N,40] = bufA @ W2 (padded to 48 cols in compute)
    k_pad_w2<<<(HID_F * OUT_P + B - 1) / B, B, 0, stream>>>(W2, W2p);
    int mtiles = N_NODES / 16;                     // 6250, exact
    k_gemm2<<<(mtiles + 7) / 8, 256, 0, stream>>>(bufA, W2p, bufB);

    // 7) layer-2 aggregation into d_out
    long n40 = (long)N_NODES * OUT_F;
    k_self_init<<<(unsigned)((n40 + B - 1) / B), B, 0, stream>>>(bufB, dinv, b2, out, OUT_F);
    long t40 = (long)NUM_E * 10;
    k_edge_scatter40<<<(unsigned)((t40 + B - 1) / B), B, 0, stream>>>(bufB, src, dst, dinv, out);
}